// ST_GCN_86242943304449
// MI455X (gfx1250) — compile-verified
//
#include <hip/hip_runtime.h>

typedef float v2f __attribute__((ext_vector_type(2)));
typedef float v8f __attribute__((ext_vector_type(8)));

#define CIN   128
#define COUT  128
#define PP    3
#define VV    4096
#define NPLANES 48          // (FIFO_SIZE-1) * P fifo planes summed
#define KB    64            // K-slab staged in LDS per step
#define KSPLIT 6            // grid.y: 3 p-values x 2 K-halves

// ---------------------------------------------------------------------------
// Kernel 1: z[r][v] = b[r] + sum_c W[r][c] * x[c][v]     (384 x 4096, fp32)
// Coalesced over v; W row is wave-uniform -> scalar loads. ~0.4 GFLOP, trivial.
// ---------------------------------------------------------------------------
__global__ void __launch_bounds__(256)
zgemm_kernel(const float* __restrict__ x, const float* __restrict__ W,
             const float* __restrict__ b, float* __restrict__ z) {
    const int v = blockIdx.x * 256 + threadIdx.x;   // 0..4095
    const int r = blockIdx.y;                       // 0..383
    const float* wr = W + r * CIN;
    float acc = b[r];
#pragma unroll 8
    for (int c = 0; c < CIN; ++c)
        acc = fmaf(wr[c], x[c * VV + v], acc);
    z[(size_t)r * VV + v] = acc;
}

// ---------------------------------------------------------------------------
// Kernel 2: epilogue bias plane: ep = xv + sum over 48 fifo planes.
// Fully coalesced streaming reduction (107 MB read once).
// ---------------------------------------------------------------------------
__global__ void __launch_bounds__(256)
fifo_sum_kernel(const float* __restrict__ fifo, const float* __restrict__ x,
                float* __restrict__ ep) {
    const int idx = blockIdx.x * 256 + threadIdx.x;   // 0..(128*4096-1)
    float s = x[idx];
#pragma unroll 8
    for (int j = 0; j < NPLANES; ++j)
        s += fifo[(size_t)j * (COUT * VV) + idx];
    ep[idx] = s;
}

// ---------------------------------------------------------------------------
// Kernel 3: partial[q] = z_p[:, kRange] @ A_p[kRange, :]  via V_WMMA_F32_16X16X4_F32
//   grid.x = 256 N-tiles (n0 = 16*bx), grid.y = 6 (p = by>>1, K-half = by&1)
//   8 waves / block, wave w owns M-tile rows [16w, 16w+16), one v8f accumulator.
//   B slab (KB x 16 of A_p) staged in LDS, shared by all 8 waves (8x reuse).
//   A fragment = contiguous float2 from z (lanes 0-15: K=k,k+1; 16-31: K=k+2,k+3).
// ---------------------------------------------------------------------------
__global__ void __launch_bounds__(256)
gemm_wmma_kernel(const float* __restrict__ z, const float* __restrict__ A,
                 float* __restrict__ part) {
    __shared__ float ldsB[KB * 16];

    const int tid  = threadIdx.x;
    const int lane = tid & 31;
    const int wid  = tid >> 5;        // 0..7 -> M tile index
    const int col  = lane & 15;
    const int half = lane >> 4;

    const int n0     = blockIdx.x * 16;
    const int p      = blockIdx.y >> 1;
    const int k0base = (blockIdx.y & 1) * (VV / 2);

    const float* __restrict__ Ap   = A + (size_t)p * VV * VV;
    // A-operand row for this lane: z[p][16*wid + (lane&15)][*]
    const float* __restrict__ zrow = z + (size_t)(p * COUT + wid * 16 + col) * VV;

    v8f acc = {};

    for (int kb = 0; kb < VV / 2; kb += KB) {
        const int k0 = k0base + kb;

        // Cooperatively stage B slab: Ap[k0+kk][n0+nn] -> ldsB[kk*16+nn]
#pragma unroll
        for (int i = 0; i < (KB * 16) / 256; ++i) {
            const int idx = tid + i * 256;
            const int kk  = idx >> 4;
            const int nn  = idx & 15;
            ldsB[idx] = Ap[(size_t)(k0 + kk) * VV + (n0 + nn)];
        }
        __syncthreads();

        // Prefetch the next slab while we compute (uniform branch, EXEC stays full)
        if (kb + KB < VV / 2)
            __builtin_prefetch(&Ap[(size_t)(k0 + KB) * VV + n0 + col], 0, 1);

#pragma unroll
        for (int k = 0; k < KB; k += 4) {
            // 16x4 f32 A fragment: VGPR0 = K (k + 2*half), VGPR1 = K (k+1 + 2*half)
            v2f afrag = *(const v2f*)(zrow + (k0 + k + 2 * half));
            // 4x16 f32 B fragment, same K split across wave halves
            v2f bfrag;
            bfrag.x = ldsB[(k +     2 * half) * 16 + col];
            bfrag.y = ldsB[(k + 1 + 2 * half) * 16 + col];
            acc = __builtin_amdgcn_wmma_f32_16x16x4_f32(
                false, afrag, false, bfrag, (short)0, acc, false, false);
        }
        __syncthreads();
    }

    // Write this K-chunk's partial plane (no atomics -> deterministic).
    float* __restrict__ out = part + (size_t)blockIdx.y * (COUT * VV);
#pragma unroll
    for (int r = 0; r < 8; ++r) {
        const int row = wid * 16 + r + 8 * half;   // C/D layout: vgpr r, wave half
        out[(size_t)row * VV + (n0 + col)] = acc[r];
    }
}

// ---------------------------------------------------------------------------
// Kernel 4: out = relu(sum of 6 partial planes + ep)
// ---------------------------------------------------------------------------
__global__ void __launch_bounds__(256)
epilogue_kernel(const float* __restrict__ part, const float* __restrict__ ep,
                float* __restrict__ out) {
    const int idx = blockIdx.x * 256 + threadIdx.x;
    float s = ep[idx];
#pragma unroll
    for (int q = 0; q < KSPLIT; ++q)
        s += part[(size_t)q * (COUT * VV) + idx];
    out[idx] = fmaxf(s, 0.0f);
}

// ---------------------------------------------------------------------------
// Launch: inputs are {x, A, fifo, W, b}; out is (1,128,4096) fp32.
// Workspace: z (384*4096) | ep (128*4096) | part (6 * 128*4096)  = 20 MB.
// ---------------------------------------------------------------------------
extern "C" void kernel_launch(void* const* d_in, const int* in_sizes, int n_in,
                              void* d_out, int out_size, void* d_ws, size_t ws_size,
                              hipStream_t stream) {
    const float* x    = (const float*)d_in[0];
    const float* A    = (const float*)d_in[1];
    const float* fifo = (const float*)d_in[2];
    const float* W    = (const float*)d_in[3];
    const float* b    = (const float*)d_in[4];
    float* out = (float*)d_out;

    float* z    = (float*)d_ws;                        // 384*4096
    float* ep   = z  + (size_t)PP * COUT * VV;         // 128*4096
    float* part = ep + (size_t)COUT * VV;              // 6 * 128*4096

    zgemm_kernel   <<<dim3(VV / 256, PP * COUT), 256, 0, stream>>>(x, W, b, z);
    fifo_sum_kernel<<<dim3((COUT * VV) / 256),   256, 0, stream>>>(fifo, x, ep);
    gemm_wmma_kernel<<<dim3(VV / 16, KSPLIT),    256, 0, stream>>>(z, A, part);
    epilogue_kernel<<<dim3((COUT * VV) / 256),   256, 0, stream>>>(part, ep, out);
}